// StrengthNet_81080392614771
// MI455X (gfx1250) — compile-verified
//
#include <hip/hip_runtime.h>

// CDNA5 / gfx1250: wave32, WMMA f32 16x16x4.
typedef __attribute__((ext_vector_type(2))) float v2f;
typedef __attribute__((ext_vector_type(8))) float v8f;

#define DHID 32
#define FDIM 6

// ---- order-preserving float<->uint encoding for atomicMax on floats ----
__device__ __forceinline__ unsigned enc_f32(float f) {
    unsigned u = __float_as_uint(f);
    return (u & 0x80000000u) ? ~u : (u | 0x80000000u);
}
__device__ __forceinline__ float dec_f32(unsigned u) {
    u = (u & 0x80000000u) ? (u & 0x7FFFFFFFu) : ~u;
    return __uint_as_float(u);
}

// ---- kernel 0: re-init per-segment scratch every call (graph-safe) ----
__global__ void init_seg(unsigned* __restrict__ m_u, float* __restrict__ denom,
                         float* __restrict__ num, int S) {
    int i = blockIdx.x * blockDim.x + threadIdx.x;
    if (i < S) { m_u[i] = 0u; denom[i] = 0.0f; num[i] = 0.0f; }
}

// ---- kernel 1: fused MLP via WMMA f32 16x16x4, transposed mapping ----
// Computes hT[32 x 16rows] = W1[32x6pad8] * xT[6pad8 x 16rows] per tile:
//   acc0 = d-rows 0..15, acc1 = d-rows 16..31.
// D layout => lane L holds h[x-row R+L%16][d = v + (L>=16?8:0)] for v=0..7.
// Head dots (r,z) reduce over d WITHIN each lane; one shfl_xor(16) merges the
// two halves; lanes 0..15 store (r,z) fully coalesced.
__global__ __launch_bounds__(256)
void mlp_rz_wmma(const float* __restrict__ x,
                 const float* __restrict__ W1, const float* __restrict__ b1,
                 const float* __restrict__ Wr, const float* __restrict__ br,
                 const float* __restrict__ Wz, const float* __restrict__ bz,
                 float2* __restrict__ rz, int N)
{
    const int  lane = threadIdx.x & 31;
    const int  l16  = lane & 15;
    const bool hiH  = lane >= 16;
    const int  kb   = hiH ? 2 : 0;     // K-pair owned by this lane half (A and B layouts)

    // ---- A tiles: W1 [32,6] row-major. A-layout 16x4 f32:
    // lane<16: VGPR0=K(kb),VGPR1=K(kb+1); lanes>=16 own K=2,3 (lo) / K=6,7->zero (hi).
    const float* w0 = W1 + l16 * FDIM;          // d-rows 0..15
    const float* w1 = W1 + (l16 + 16) * FDIM;   // d-rows 16..31
    v2f A0lo, A0hi, A1lo, A1hi;
    A0lo.x = w0[kb];  A0lo.y = w0[kb + 1];
    A1lo.x = w1[kb];  A1lo.y = w1[kb + 1];
    A0hi.x = hiH ? 0.0f : w0[4];  A0hi.y = hiH ? 0.0f : w0[5];
    A1hi.x = hiH ? 0.0f : w1[4];  A1hi.y = hiH ? 0.0f : w1[5];

    // ---- per-lane epilogue constants: this lane's 16 hidden dims ----
    const int dbase = hiH ? 8 : 0;
    float b1a[8], b1b[8], wra[8], wrb[8], wza[8], wzb[8];
    #pragma unroll
    for (int v = 0; v < 8; ++v) {
        b1a[v] = b1[dbase + v];       b1b[v] = b1[dbase + v + 16];
        wra[v] = Wr[dbase + v];       wrb[v] = Wr[dbase + v + 16];
        wza[v] = Wz[dbase + v];       wzb[v] = Wz[dbase + v + 16];
    }
    const float brS = br[0], bzS = bz[0];
    const float hmask = hiH ? 0.0f : 1.0f;   // zero K=6,7 of B without a branch

    const int wave   = (blockIdx.x * blockDim.x + threadIdx.x) >> 5;
    const int nWaves = (gridDim.x * blockDim.x) >> 5;
    const int nTiles = N >> 4;

    for (int t = wave; t < nTiles; t += nWaves) {
        const int R   = t << 4;
        const int row = R + l16;
        const float* xr = x + row * FDIM;

        // B tile = xT chunk. B-layout 4x16: VGPR v holds K=kb+v (this lane half).
        // row*6+kb and row*6+4 are even dword offsets -> 8B-aligned float2 loads.
        v2f Blo = *(const v2f*)(xr + kb);
        v2f Bt  = *(const v2f*)(xr + 4);       // unconditional (valid addr all lanes)
        v2f Bhi; Bhi.x = Bt.x * hmask; Bhi.y = Bt.y * hmask;   // K=6,7 -> 0

        v8f acc0 = {};
        v8f acc1 = {};
        acc0 = __builtin_amdgcn_wmma_f32_16x16x4_f32(false, A0lo, false, Blo, (short)0, acc0, false, false);
        acc1 = __builtin_amdgcn_wmma_f32_16x16x4_f32(false, A1lo, false, Blo, (short)0, acc1, false, false);
        acc0 = __builtin_amdgcn_wmma_f32_16x16x4_f32(false, A0hi, false, Bhi, (short)0, acc0, false, false);
        acc1 = __builtin_amdgcn_wmma_f32_16x16x4_f32(false, A1hi, false, Bhi, (short)0, acc1, false, false);

        // Per-lane epilogue: bias + ReLU + r/z partial dots over this lane's 16 dims.
        float rr = 0.0f, zz = 0.0f;
        #pragma unroll
        for (int v = 0; v < 8; ++v) {
            float h0 = fmaxf(acc0[v] + b1a[v], 0.0f);
            float h1 = fmaxf(acc1[v] + b1b[v], 0.0f);
            rr += h0 * wra[v] + h1 * wrb[v];
            zz += h0 * wza[v] + h1 * wzb[v];
        }
        // merge the two lane halves (d 0..7,16..23 vs 8..15,24..31 split)
        rr += __shfl_xor(rr, 16, 32);
        zz += __shfl_xor(zz, 16, 32);

        if (!hiH) {   // lanes 0..15: coalesced 128B store of (r,z) for 16 rows
            rz[R + l16] = make_float2(rr + brS, zz + bzS);
        }
    }
}

// ---- kernel 2: segmented max of z (sorted seg ids -> chunked, few atomics) ----
__global__ void seg_max_kernel(const float2* __restrict__ rz, const int* __restrict__ seg,
                               unsigned* __restrict__ m_u, int N)
{
    const int C = 16;
    int t  = blockIdx.x * blockDim.x + threadIdx.x;
    int i0 = t * C;
    if (i0 >= N) return;
    int end = i0 + C; if (end > N) end = N;

    int   cur = seg[i0];
    float mx  = rz[i0].y;
    for (int i = i0 + 1; i < end; ++i) {
        int   s  = seg[i];
        float zv = rz[i].y;
        if (s == cur) { mx = fmaxf(mx, zv); }
        else { atomicMax(&m_u[cur], enc_f32(mx)); cur = s; mx = zv; }
    }
    atomicMax(&m_u[cur], enc_f32(mx));
}

// ---- kernel 3: segmented exp-sum and weighted sum (chunked atomics) ----
__global__ void seg_expsum_kernel(const float2* __restrict__ rz, const int* __restrict__ seg,
                                  const unsigned* __restrict__ m_u,
                                  float* __restrict__ denom, float* __restrict__ num, int N)
{
    const int C = 8;
    int t  = blockIdx.x * blockDim.x + threadIdx.x;
    int i0 = t * C;
    if (i0 >= N) return;
    int end = i0 + C; if (end > N) end = N;

    int   cur = seg[i0];
    float m   = dec_f32(m_u[cur]);
    float de = 0.0f, nu = 0.0f;
    for (int i = i0; i < end; ++i) {
        int s = seg[i];
        if (s != cur) {
            atomicAdd(&denom[cur], de);
            atomicAdd(&num[cur],   nu);
            cur = s; m = dec_f32(m_u[cur]); de = 0.0f; nu = 0.0f;
        }
        float2 v = rz[i];
        float  e = __expf(v.y - m);
        de += e;
        nu += e * v.x;
    }
    atomicAdd(&denom[cur], de);
    atomicAdd(&num[cur],   nu);
}

// ---- kernel 4: finalize ----
__global__ void finalize_pred(const float* __restrict__ num, const float* __restrict__ denom,
                              float* __restrict__ pred, int S)
{
    int i = blockIdx.x * blockDim.x + threadIdx.x;
    if (i < S) pred[i] = num[i] / denom[i];
}

extern "C" void kernel_launch(void* const* d_in, const int* in_sizes, int n_in,
                              void* d_out, int out_size, void* d_ws, size_t ws_size,
                              hipStream_t stream)
{
    const float* x   = (const float*)d_in[0];
    const int*   seg = (const int*)  d_in[1];
    // d_in[2] = num_segments scalar on device; use out_size instead (== S)
    const float* W1  = (const float*)d_in[3];
    const float* b1  = (const float*)d_in[4];
    const float* Wr  = (const float*)d_in[5];
    const float* br  = (const float*)d_in[6];
    const float* Wz  = (const float*)d_in[7];
    const float* bz  = (const float*)d_in[8];

    const int N = in_sizes[1];   // one seg id per row
    const int S = out_size;
    float* pred = (float*)d_out;

    // workspace layout: [ float2 rz[N] | unsigned m_u[S] | float denom[S] | float num[S] ]
    float2*   rz    = (float2*)d_ws;
    unsigned* m_u   = (unsigned*)((char*)d_ws + (size_t)N * sizeof(float2));
    float*    denom = (float*)(m_u + S);
    float*    num   = denom + S;

    init_seg<<<(S + 255) / 256, 256, 0, stream>>>(m_u, denom, num, S);

    // ~8192 waves streaming 16-row tiles (N/16 = 131072 tiles -> 16 tiles/wave)
    mlp_rz_wmma<<<1024, 256, 0, stream>>>(x, W1, b1, Wr, br, Wz, bz, rz, N);

    int t1 = (N + 15) / 16;
    seg_max_kernel<<<(t1 + 255) / 256, 256, 0, stream>>>(rz, seg, m_u, N);

    int t2 = (N + 7) / 8;
    seg_expsum_kernel<<<(t2 + 255) / 256, 256, 0, stream>>>(rz, seg, m_u, denom, num, N);

    finalize_pred<<<(S + 255) / 256, 256, 0, stream>>>(num, denom, pred, S);
}